// TeamGraphSAGE_86577950752953
// MI455X (gfx1250) — compile-verified
//
#include <hip/hip_runtime.h>

#define N_NODES 50000
#define KDIM    256      // inner dim for both SAGE layers and the decoder cat
#define D_OUT   128

typedef __attribute__((ext_vector_type(2))) float v2f;
typedef __attribute__((ext_vector_type(8))) float v8f;

// ---------------------------------------------------------------------------
// 1) degree histogram (int): deg[dst[e]] += 1
// ---------------------------------------------------------------------------
__global__ void deg_kernel(const int* __restrict__ dst, int* __restrict__ deg, int E) {
    int e = blockIdx.x * blockDim.x + threadIdx.x;
    if (e < E) atomicAdd(&deg[dst[e]], 1);
}

// ---------------------------------------------------------------------------
// 2) hierarchical exclusive scan over deg[0..n) -> offs
// ---------------------------------------------------------------------------
__global__ void __launch_bounds__(256)
scan_part(const int* __restrict__ deg, int* __restrict__ offs,
          int* __restrict__ blockSums, int n) {
    __shared__ int tmp[256];
    const int tid = threadIdx.x;
    const int gid = blockIdx.x * 256 + tid;
    const int v = (gid < n) ? deg[gid] : 0;
    tmp[tid] = v;
    __syncthreads();
    #pragma unroll
    for (int d = 1; d < 256; d <<= 1) {
        int t = (tid >= d) ? tmp[tid - d] : 0;
        __syncthreads();
        tmp[tid] += t;
        __syncthreads();
    }
    const int incl = tmp[tid];
    if (gid < n) offs[gid] = incl - v;                  // exclusive
    if (tid == 255) blockSums[blockIdx.x] = incl;
}

__global__ void __launch_bounds__(256)
scan_sums(int* __restrict__ blockSums, int nb) {        // nb <= 256, one block
    __shared__ int tmp[256];
    const int tid = threadIdx.x;
    const int v = (tid < nb) ? blockSums[tid] : 0;
    tmp[tid] = v;
    __syncthreads();
    #pragma unroll
    for (int d = 1; d < 256; d <<= 1) {
        int t = (tid >= d) ? tmp[tid - d] : 0;
        __syncthreads();
        tmp[tid] += t;
        __syncthreads();
    }
    if (tid < nb) blockSums[tid] = tmp[tid] - v;        // exclusive
}

__global__ void __launch_bounds__(256)
scan_add(int* __restrict__ offs, const int* __restrict__ blockSums, int n, int E) {
    const int gid = blockIdx.x * 256 + threadIdx.x;
    if (gid < n) offs[gid] += blockSums[blockIdx.x];
    if (gid == 0) offs[n] = E;
}

// ---------------------------------------------------------------------------
// 3) bucket edges into CSR: sorted_src[pos] = src[e], pos from cursor[dst[e]]
// ---------------------------------------------------------------------------
__global__ void fill_csr(const int* __restrict__ src, const int* __restrict__ dst,
                         int* __restrict__ cursor, int* __restrict__ sorted_src, int E) {
    int e = blockIdx.x * blockDim.x + threadIdx.x;
    if (e < E) {
        int pos = atomicAdd(&cursor[dst[e]], 1);
        sorted_src[pos] = src[e];
    }
}

// ---------------------------------------------------------------------------
// 4) segment mean: out[i,:] = mean_{e in seg(i)} feat[sorted_src[e],:]
//    one wave per node; lane owns 8 contiguous features (wave reads 1KB rows
//    fully coalesced; rows live in L2 after first touch).
// ---------------------------------------------------------------------------
__global__ void __launch_bounds__(256)
aggregate_mean(const float* __restrict__ feat, const int* __restrict__ offs,
               const int* __restrict__ sorted_src, float* __restrict__ out, int n) {
    const int node = blockIdx.x * 8 + (threadIdx.x >> 5);
    if (node >= n) return;
    const int lane = threadIdx.x & 31;
    const int beg = offs[node];
    const int end = offs[node + 1];

    float a0 = 0.f, a1 = 0.f, a2 = 0.f, a3 = 0.f;
    float a4 = 0.f, a5 = 0.f, a6 = 0.f, a7 = 0.f;
    for (int e = beg; e < end; ++e) {
        const float* row = feat + (long long)sorted_src[e] * KDIM + lane * 8;
        if (e + 1 < end)
            __builtin_prefetch(feat + (long long)sorted_src[e + 1] * KDIM + lane * 8, 0, 1);
        float4 v0 = *(const float4*)(row);
        float4 v1 = *(const float4*)(row + 4);
        a0 += v0.x; a1 += v0.y; a2 += v0.z; a3 += v0.w;
        a4 += v1.x; a5 += v1.y; a6 += v1.z; a7 += v1.w;
    }
    const float inv = 1.0f / fmaxf((float)(end - beg), 1.0f);
    float* o = out + (long long)node * KDIM + lane * 8;
    float4 r0 = {a0 * inv, a1 * inv, a2 * inv, a3 * inv};
    float4 r1 = {a4 * inv, a5 * inv, a6 * inv, a7 * inv};
    *(float4*)(o)     = r0;
    *(float4*)(o + 4) = r1;
}

// ---------------------------------------------------------------------------
// 5) fused SAGE GEMM: out = act( Aagg @ Wl^T + X @ Wr^T + bias )
//    one wave -> one 16x16 tile; Wl^T/Wr^T 16-col panels staged in LDS.
// ---------------------------------------------------------------------------
__global__ void __launch_bounds__(256)
sage_gemm_kernel(const float* __restrict__ Aagg, const float* __restrict__ X,
                 const float* __restrict__ Wl, const float* __restrict__ Wr,
                 const float* __restrict__ bias,
                 float* __restrict__ out, int n_rows, int n_cols, int do_relu) {
    __shared__ float Bl[KDIM * 16];
    __shared__ float Br[KDIM * 16];

    const int tid     = threadIdx.x;
    const int colbase = blockIdx.x * 16;

    for (int idx = tid; idx < KDIM * 16; idx += 256) {
        int n = idx >> 8;            // 0..15
        int k = idx & (KDIM - 1);    // consecutive tid -> consecutive k (coalesced)
        Bl[k * 16 + n] = Wl[(colbase + n) * KDIM + k];
        Br[k * 16 + n] = Wr[(colbase + n) * KDIM + k];
    }
    __syncthreads();

    const int lane    = tid & 31;
    const int wave    = tid >> 5;
    const int tileRow = blockIdx.y * 8 + wave;
    if (tileRow * 16 >= n_rows) return;                // uniform per wave

    const int m     = lane & 15;
    const int khalf = (lane >> 4) << 1;                // 0 or 2
    const int row   = tileRow * 16 + m;

    const float* arow = Aagg + (long long)row * KDIM;
    const float* xrow = X    + (long long)row * KDIM;

    v8f acc = {0.f, 0.f, 0.f, 0.f, 0.f, 0.f, 0.f, 0.f};

    #pragma unroll 8
    for (int k = 0; k < KDIM; k += 4) {
        const int k0 = k + khalf;
        v2f a = *(const v2f*)(arow + k0);
        v2f b;
        b.x = Bl[k0 * 16 + m];
        b.y = Bl[(k0 + 1) * 16 + m];
        acc = __builtin_amdgcn_wmma_f32_16x16x4_f32(false, a, false, b,
                                                    (short)0, acc, false, false);
    }
    #pragma unroll 8
    for (int k = 0; k < KDIM; k += 4) {
        const int k0 = k + khalf;
        v2f a = *(const v2f*)(xrow + k0);
        v2f b;
        b.x = Br[k0 * 16 + m];
        b.y = Br[(k0 + 1) * 16 + m];
        acc = __builtin_amdgcn_wmma_f32_16x16x4_f32(false, a, false, b,
                                                    (short)0, acc, false, false);
    }

    const float bn        = bias[colbase + m];
    const int  rowOutBase = tileRow * 16 + ((lane >> 4) << 3);
    #pragma unroll
    for (int r = 0; r < 8; ++r) {
        float v = acc[r] + bn;
        if (do_relu) v = fmaxf(v, 0.0f);
        out[(long long)(rowOutBase + r) * n_cols + colbase + m] = v;
    }
}

// ---------------------------------------------------------------------------
// 6) decoder: per wave, 16 edges.  A[m,:] = [z[src[e_m]] | z[dst[e_m]]],
//    B = We1^T (256x16, LDS). acc -> hidden; relu, dot We2, 16-lane reduce.
// ---------------------------------------------------------------------------
__global__ void __launch_bounds__(256)
decode_kernel(const float* __restrict__ z,
              const int* __restrict__ src, const int* __restrict__ dst,
              const float* __restrict__ We1, const float* __restrict__ be1,
              const float* __restrict__ We2, const float* __restrict__ be2,
              float* __restrict__ logits, int E) {
    __shared__ float Bw[KDIM * 16];
    const int tid = threadIdx.x;
    for (int idx = tid; idx < KDIM * 16; idx += 256) {
        int j = idx >> 8;
        int k = idx & (KDIM - 1);
        Bw[k * 16 + j] = We1[j * KDIM + k];
    }
    __syncthreads();

    const int lane  = tid & 31;
    const int wave  = tid >> 5;
    const long long ebase = ((long long)blockIdx.x * 8 + wave) * 16;
    if (ebase >= E) return;

    const int m     = lane & 15;
    const int khalf = (lane >> 4) << 1;
    const long long e = ebase + m;
    const float* zs = z + (long long)src[e] * D_OUT;
    const float* zd = z + (long long)dst[e] * D_OUT;

    v8f acc = {0.f, 0.f, 0.f, 0.f, 0.f, 0.f, 0.f, 0.f};

    #pragma unroll 8
    for (int k = 0; k < 128; k += 4) {
        const int k0 = k + khalf;
        v2f a = *(const v2f*)(zs + k0);
        v2f b;
        b.x = Bw[k0 * 16 + m];
        b.y = Bw[(k0 + 1) * 16 + m];
        acc = __builtin_amdgcn_wmma_f32_16x16x4_f32(false, a, false, b,
                                                    (short)0, acc, false, false);
    }
    #pragma unroll 8
    for (int k = 128; k < 256; k += 4) {
        const int k0 = k + khalf;
        v2f a = *(const v2f*)(zd + (k0 - 128));
        v2f b;
        b.x = Bw[k0 * 16 + m];
        b.y = Bw[(k0 + 1) * 16 + m];
        acc = __builtin_amdgcn_wmma_f32_16x16x4_f32(false, a, false, b,
                                                    (short)0, acc, false, false);
    }

    const float bj  = be1[m];
    const float wj  = We2[m];
    const float b2v = be2[0];
    #pragma unroll
    for (int r = 0; r < 8; ++r) {
        float p = fmaxf(acc[r] + bj, 0.0f) * wj;
        p += __shfl_xor(p, 1);
        p += __shfl_xor(p, 2);
        p += __shfl_xor(p, 4);
        p += __shfl_xor(p, 8);          // stays within each 16-lane half
        if (m == 0) {
            long long eo = ebase + r + ((lane >> 4) << 3);
            logits[eo] = p + b2v;
        }
    }
}

// ---------------------------------------------------------------------------
extern "C" void kernel_launch(void* const* d_in, const int* in_sizes, int n_in,
                              void* d_out, int out_size, void* d_ws, size_t ws_size,
                              hipStream_t stream) {
    const float* x   = (const float*)d_in[0];
    const int*   ei  = (const int*)d_in[1];
    const float* W1l = (const float*)d_in[2];
    const float* b1  = (const float*)d_in[3];
    const float* W1r = (const float*)d_in[4];
    const float* W2l = (const float*)d_in[5];
    const float* b2  = (const float*)d_in[6];
    const float* W2r = (const float*)d_in[7];
    const float* We1 = (const float*)d_in[8];
    const float* be1 = (const float*)d_in[9];
    const float* We2 = (const float*)d_in[10];
    const float* be2 = (const float*)d_in[11];
    float* out = (float*)d_out;

    const int E = in_sizes[1] / 2;
    const int* src = ei;
    const int* dst = ei + E;

    // workspace carve-out (256B aligned slabs)
    auto align256 = [](size_t v) { return (v + 255) & ~(size_t)255; };
    char*  ws  = (char*)d_ws;
    size_t off = 0;
    int* deg        = (int*)(ws + off); off += align256((size_t)N_NODES * 4);
    int* offs       = (int*)(ws + off); off += align256((size_t)(N_NODES + 1) * 4);
    int* cursor     = (int*)(ws + off); off += align256((size_t)N_NODES * 4);
    int* blockSums  = (int*)(ws + off); off += align256(256 * 4);
    int* sorted_src = (int*)(ws + off); off += align256((size_t)E * 4);
    float* sbuf = (float*)(ws + off); off += (size_t)N_NODES * KDIM * 4;   // 51.2 MB
    float* hbuf = (float*)(ws + off); off += (size_t)N_NODES * KDIM * 4;   // 51.2 MB
    float* zbuf = (float*)(ws + off);                                      // 25.6 MB

    const int NB = (N_NODES + 255) / 256;              // 196 scan chunks
    const int rowTiles = (N_NODES + 15) / 16;          // 3125
    const int gy = (rowTiles + 7) / 8;                 // 391
    const int aggBlocks = (N_NODES + 7) / 8;           // wave per node, 8 waves/block

    // ---- build CSR (once; shared by both layers) ----
    hipMemsetAsync(deg, 0, (size_t)N_NODES * 4, stream);
    deg_kernel<<<(E + 255) / 256, 256, 0, stream>>>(dst, deg, E);
    scan_part<<<NB, 256, 0, stream>>>(deg, offs, blockSums, N_NODES);
    scan_sums<<<1, 256, 0, stream>>>(blockSums, NB);
    scan_add<<<NB, 256, 0, stream>>>(offs, blockSums, N_NODES, E);
    hipMemcpyAsync(cursor, offs, (size_t)N_NODES * 4, hipMemcpyDeviceToDevice, stream);
    fill_csr<<<(E + 255) / 256, 256, 0, stream>>>(src, dst, cursor, sorted_src, E);

    // ---- layer 1 ----
    aggregate_mean<<<aggBlocks, 256, 0, stream>>>(x, offs, sorted_src, sbuf, N_NODES);
    sage_gemm_kernel<<<dim3(256 / 16, gy), 256, 0, stream>>>(
        sbuf, x, W1l, W1r, b1, hbuf, N_NODES, 256, 1);

    // ---- layer 2 ----
    aggregate_mean<<<aggBlocks, 256, 0, stream>>>(hbuf, offs, sorted_src, sbuf, N_NODES);
    sage_gemm_kernel<<<dim3(D_OUT / 16, gy), 256, 0, stream>>>(
        sbuf, hbuf, W2l, W2r, b2, zbuf, N_NODES, D_OUT, 0);

    // ---- decoder ----
    const int waves = E / 16;                          // 50000
    decode_kernel<<<(waves + 7) / 8, 256, 0, stream>>>(
        zbuf, src, dst, We1, be1, We2, be2, out, E);
}